// LongRankIC_11330123727500
// MI455X (gfx1250) — compile-verified
//
#include <hip/hip_runtime.h>
#include <stdint.h>

// Problem constants (preds/targets: (8192, 64, 64) f32 -> N=8192 rows, C=4096 cols)
#define NTHREADS 512
#define N_ROWS   8192
#define N_COLS   4096
#define K_TOP    4096   // int(N * 0.5)

// LDS layout (dynamic shared, 132 KB total):
//   [0,      32KB)  : float pcol[8192]           (TDM target)
//   [32KB,   64KB)  : float tcol[8192]           (TDM target)
//   [64KB,  128KB)  : u64  keys[8192]            (bitonic sort buffer; 2nd sort reuses top half)
//   [128KB, 132KB)  : i64  red[NTHREADS]         (reduction scratch)
#define SMEM_BYTES (132 * 1024)

typedef unsigned int v4u __attribute__((ext_vector_type(4)));
typedef int          v8i __attribute__((ext_vector_type(8)));
typedef int          v4i __attribute__((ext_vector_type(4)));

#if defined(__has_builtin)
#if __has_builtin(__builtin_amdgcn_tensor_load_to_lds)
#define HAVE_TDM 1
#endif
#endif

// Monotonic float->uint mapping: a < b  <=>  fo(a) < fo(b)
__device__ __forceinline__ unsigned f32_orderable(float f) {
    unsigned u = __float_as_uint(f);
    return (u & 0x80000000u) ? ~u : (u | 0x80000000u);
}

#ifdef HAVE_TDM
// DMA one strided column (tile 1 x N_ROWS, row stride N_COLS elements) into LDS.
// D# per CDNA5 ISA: group0 = {ctrl, lds_addr, gaddr_lo, gaddr_hi|type},
// group1 packs data_size/dims/tile/strides; groups 2&3 zero (2D tensor, 2D tile).
// This toolchain's builtin is the 6-arg form:
//   (uint32x4 g0, int32x8 g1, int32x4 g2, int32x4 g3, int32x8 pad, i32 cpol)
__device__ __forceinline__ void tdm_load_column(const float* base, int col, unsigned lds_byte_off) {
    unsigned long long ga = (unsigned long long)(uintptr_t)(base + col);
    v4u g0;
    g0[0] = 1u;                                    // count=1 (valid), user mode, no gather
    g0[1] = lds_byte_off;                          // lds_addr (bytes)
    g0[2] = (unsigned)ga;                          // global_addr[31:0]
    g0[3] = (unsigned)(ga >> 32) | (2u << 30);     // global_addr[56:32] | type=2 ("image")

    // group1, packed as 4 x u64 (bits 0..255)
    unsigned long long q0 = (2ull << 16)                                  // data_size = 2 (4 bytes)
                          | ((unsigned long long)(N_COLS & 0xffff) << 48);// tensor_dim0 low16 (bits 48..63)
    unsigned long long q1 = ((unsigned long long)((N_COLS >> 16) & 0xffff))   // tensor_dim0 hi16 (bits 64..79)
                          | ((unsigned long long)N_ROWS << 16)                // tensor_dim1 (bits 80..111)
                          | (1ull << 48);                                     // tile_dim0 = 1 (bits 112..127)
    unsigned long long q2 = (unsigned long long)N_ROWS                        // tile_dim1 = 8192 (bits 128..143)
                          /* tile_dim2 = 0 (unused) */
                          | ((unsigned long long)N_COLS << 32);               // tensor_dim0_stride lo32 (bits 160..191)
    unsigned long long q3 = 0ull;                                             // stride hi bits, tensor_dim1_stride = 0

    v8i g1;
    g1[0] = (int)(unsigned)q0; g1[1] = (int)(unsigned)(q0 >> 32);
    g1[2] = (int)(unsigned)q1; g1[3] = (int)(unsigned)(q1 >> 32);
    g1[4] = (int)(unsigned)q2; g1[5] = (int)(unsigned)(q2 >> 32);
    g1[6] = (int)(unsigned)q3; g1[7] = (int)(unsigned)(q3 >> 32);
    v4i z4 = {0, 0, 0, 0};
    v8i z8 = {0, 0, 0, 0, 0, 0, 0, 0};
    __builtin_amdgcn_tensor_load_to_lds(g0, g1, z4, z4, z8, 0);
}
#endif

extern "C" __global__ void __launch_bounds__(NTHREADS)
rank_ic_kernel(const float* __restrict__ preds,
               const float* __restrict__ targets,
               float* __restrict__ out) {
    extern __shared__ unsigned char smem_raw[];
    float*              pcol = (float*)(smem_raw);
    float*              tcol = (float*)(smem_raw + 32 * 1024);
    unsigned long long* keys = (unsigned long long*)(smem_raw + 64 * 1024);
    long long*          red  = (long long*)(smem_raw + 128 * 1024);

    const int tid = (int)threadIdx.x;
    const int col = (int)blockIdx.x;

#ifdef HAVE_TDM
    // One wave issues both column DMAs; TENSORcnt is per-wave, so it alone waits,
    // then the workgroup barrier makes the LDS tiles visible to everyone.
    if (tid == 0) {
        unsigned smem_base = (unsigned)(uintptr_t)smem_raw;   // flat addr[31:0] == LDS offset
        tdm_load_column(preds,   col, smem_base);
        tdm_load_column(targets, col, smem_base + 32 * 1024);
        __builtin_amdgcn_s_wait_tensorcnt(0);
    }
#else
    for (int i = tid; i < N_ROWS; i += NTHREADS) {
        pcol[i] = preds[(size_t)i * N_COLS + col];
        tcol[i] = targets[(size_t)i * N_COLS + col];
    }
#endif
    __syncthreads();

    // ---- Pass 1: sort (p, idx) descending. keys = orderable(p)<<32 | row ----
    for (int i = tid; i < N_ROWS; i += NTHREADS)
        keys[i] = ((unsigned long long)f32_orderable(pcol[i]) << 32) | (unsigned)i;
    __syncthreads();

    for (int size = 2; size <= N_ROWS; size <<= 1) {
        for (int stride = size >> 1; stride > 0; stride >>= 1) {
            for (int t = tid; t < (N_ROWS >> 1); t += NTHREADS) {
                int i = 2 * t - (t & (stride - 1));
                int j = i + stride;
                bool up = ((i & size) != 0);            // overall DESCENDING sort
                unsigned long long x = keys[i], y = keys[j];
                if ((x > y) == up) { keys[i] = y; keys[j] = x; }
            }
            __syncthreads();
        }
    }

    // ---- Gather targets of the top-k and build (t, pos) keys in upper half ----
    unsigned long long* k2 = keys + K_TOP;
    for (int i = tid; i < K_TOP; i += NTHREADS) {
        unsigned row = (unsigned)keys[i] & (N_ROWS - 1);    // original row index
        k2[i] = ((unsigned long long)f32_orderable(tcol[row]) << 32) | (unsigned)i;
    }
    __syncthreads();

    // ---- Pass 2: sort (t, pos) ascending over K_TOP elements ----
    for (int size = 2; size <= K_TOP; size <<= 1) {
        for (int stride = size >> 1; stride > 0; stride >>= 1) {
            for (int t = tid; t < (K_TOP >> 1); t += NTHREADS) {
                int i = 2 * t - (t & (stride - 1));
                int j = i + stride;
                bool up = ((i & size) == 0);            // overall ASCENDING sort
                unsigned long long x = k2[i], y = k2[j];
                if ((x > y) == up) { k2[i] = y; k2[j] = x; }
            }
            __syncthreads();
        }
    }

    // ---- S = sum_j p_r[pos(j)] * j, with p_r[i] = K-1-i; exact in int64 ----
    long long part = 0;
    for (int j = tid; j < K_TOP; j += NTHREADS) {
        int i = (int)((unsigned)k2[j] & (K_TOP - 1));   // position in descending-p order
        part += (long long)(K_TOP - 1 - i) * (long long)j;
    }
    red[tid] = part;
    __syncthreads();
    for (int s = NTHREADS >> 1; s > 0; s >>= 1) {
        if (tid < s) red[tid] += red[tid + s];
        __syncthreads();
    }

    if (tid == 0) {
        double S  = (double)red[0];
        double m  = (double)(K_TOP - 1) * 0.5;
        double cov = S / (double)K_TOP - m * m;
        double stdprod = (double)K_TOP * (double)(K_TOP + 1) / 12.0; // std(p_r)*std(t_r), ddof=1
        out[col] = (float)(cov / (stdprod + 1e-8));
    }
}

extern "C" void kernel_launch(void* const* d_in, const int* in_sizes, int n_in,
                              void* d_out, int out_size, void* d_ws, size_t ws_size,
                              hipStream_t stream) {
    (void)in_sizes; (void)n_in; (void)d_ws; (void)ws_size;
    const float* preds   = (const float*)d_in[0];
    const float* targets = (const float*)d_in[1];
    float*       out     = (float*)d_out;

    // Allow >64KB dynamic LDS (deterministic, graph-capture safe).
    (void)hipFuncSetAttribute((const void*)rank_ic_kernel,
                              hipFuncAttributeMaxDynamicSharedMemorySize, SMEM_BYTES);

    dim3 grid((unsigned)out_size);   // 4096 columns, one workgroup each
    dim3 block(NTHREADS);
    hipLaunchKernelGGL(rank_ic_kernel, grid, block, SMEM_BYTES, stream,
                       preds, targets, out);
}